// ChamferLoss_82214263980148
// MI455X (gfx1250) — compile-verified
//
#include <hip/hip_runtime.h>
#include <math.h>

typedef __attribute__((ext_vector_type(2))) float v2f;
typedef __attribute__((ext_vector_type(8))) float v8f;

#define WAVES   8
#define ATILES  2
#define ROWS_PER_BLOCK (WAVES * ATILES * 16)   // 256

// For each row n of P (per batch), compute min over all rows m of T of
// squared Euclidean distance:  rowmin[b][n] = max(|p|^2 + min_m(|t|^2 - 2 p.t), 0).
// Cross term via V_WMMA_F32_16X16X4_F32:
//   A row n = (px, py, pz, 1),  B col m = (-2tx, -2ty, -2tz, |t|^2)
// The row-constant |p|^2 and the clamp commute with the min over m, so they
// are applied once per row AFTER the sweep -> hot loop is wmma + v_min only.
__global__ __launch_bounds__(256) void chamfer_rowmin_kernel(
    const float* __restrict__ P,   // [B, N, 3]
    const float* __restrict__ T,   // [B, M, 3]
    float* __restrict__ rowmin,    // [B, N]
    int N, int M)
{
    const int b    = blockIdx.y;
    const int lane = threadIdx.x & 31;
    const int wave = threadIdx.x >> 5;
    const int half = lane >> 4;      // 0: K={0,1}, 1: K={2,3}
    const int l15  = lane & 15;

    const float* __restrict__ Pb = P + (size_t)b * N * 3;
    const float* __restrict__ Tb = T + (size_t)b * M * 3;

    // ---- Build A tiles (16x4 f32 each: 2 VGPRs) and per-lane |p|^2 gather ----
    v2f   a[ATILES];
    float sqv[ATILES][8];
    int   rowbase[ATILES];
    #pragma unroll
    for (int t = 0; t < ATILES; ++t) {
        const int rb = (blockIdx.x * WAVES + wave) * (16 * ATILES) + t * 16;
        rowbase[t] = rb;
        const int r = rb + l15;
        const float px = Pb[(size_t)r * 3 + 0];
        const float py = Pb[(size_t)r * 3 + 1];
        const float pz = Pb[(size_t)r * 3 + 2];
        // A VGPR layout (16x4 f32): lanes 0-15 hold K=0 (v0), K=1 (v1);
        // lanes 16-31 hold K=2 (v0), K=3 (v1); row M = lane & 15.
        a[t].x = half ? pz   : px;
        a[t].y = half ? 1.0f : py;
        const float sq = px * px + py * py + pz * pz;
        // C/D layout: VGPR j holds row (j + 8*half) -> gather those |p|^2 values.
        #pragma unroll
        for (int j = 0; j < 8; ++j)
            sqv[t][j] = __shfl(sq, half * 8 + j, 32);
    }

    float acc[ATILES][8];
    #pragma unroll
    for (int t = 0; t < ATILES; ++t)
        #pragma unroll
        for (int j = 0; j < 8; ++j)
            acc[t][j] = INFINITY;

    // ---- Sweep all M columns, 16 at a time (2 tiles in flight) ----
    #pragma unroll 2
    for (int m0 = 0; m0 < M; m0 += 16) {
        const int c = m0 + l15;
        const float tx = Tb[(size_t)c * 3 + 0];
        const float ty = Tb[(size_t)c * 3 + 1];
        const float tz = Tb[(size_t)c * 3 + 2];
        const float sqt = tx * tx + ty * ty + tz * tz;
        // B VGPR layout (4x16 f32): lanes 0-15 hold K=0 (v0), K=1 (v1);
        // lanes 16-31 hold K=2 (v0), K=3 (v1); col N = lane & 15.
        v2f bb;
        bb.x = half ? (-2.0f * tz) : (-2.0f * tx);
        bb.y = half ? sqt          : (-2.0f * ty);

        #pragma unroll
        for (int t = 0; t < ATILES; ++t) {
            v8f c8 = {};
            c8 = __builtin_amdgcn_wmma_f32_16x16x4_f32(
                /*neg_a=*/false, a[t], /*neg_b=*/false, bb,
                /*c_mod=*/(short)0, c8, /*reuse_a=*/false, /*reuse_b=*/false);
            #pragma unroll
            for (int j = 0; j < 8; ++j)
                acc[t][j] = fminf(acc[t][j], c8[j]);   // min of |t|^2 - 2 p.t
        }
    }

    // ---- Min across the 16 column-lanes, add |p|^2, clamp, store row mins ----
    #pragma unroll
    for (int t = 0; t < ATILES; ++t) {
        #pragma unroll
        for (int j = 0; j < 8; ++j) {
            float v = acc[t][j];
            v = fminf(v, __shfl_xor(v, 1, 32));
            v = fminf(v, __shfl_xor(v, 2, 32));
            v = fminf(v, __shfl_xor(v, 4, 32));
            v = fminf(v, __shfl_xor(v, 8, 32));
            if (l15 == 0)
                rowmin[(size_t)b * N + rowbase[t] + half * 8 + j] =
                    fmaxf(v + sqv[t][j], 0.0f);
        }
    }
}

// chamfer[b] = 0.5 * ( mean_n sqrt(rowmin_P[b,n]) + mean_m sqrt(rowmin_T[b,m]) )
__global__ __launch_bounds__(256) void chamfer_reduce_kernel(
    const float* __restrict__ rmP, const float* __restrict__ rmT,
    float* __restrict__ out, int N, int M)
{
    const int b = blockIdx.x;
    __shared__ float smem[256];
    float sp = 0.0f, st = 0.0f;
    for (int i = threadIdx.x; i < N; i += 256) sp += sqrtf(rmP[(size_t)b * N + i]);
    for (int i = threadIdx.x; i < M; i += 256) st += sqrtf(rmT[(size_t)b * M + i]);
    smem[threadIdx.x] = sp / (float)N + st / (float)M;
    __syncthreads();
    for (int o = 128; o > 0; o >>= 1) {
        if (threadIdx.x < o) smem[threadIdx.x] += smem[threadIdx.x + o];
        __syncthreads();
    }
    if (threadIdx.x == 0) out[b] = smem[0] * 0.5f;
}

extern "C" void kernel_launch(void* const* d_in, const int* in_sizes, int n_in,
                              void* d_out, int out_size, void* d_ws, size_t ws_size,
                              hipStream_t stream) {
    const float* pred = (const float*)d_in[0];   // [B, N, 3]
    const float* tgt  = (const float*)d_in[1];   // [B, M, 3]
    float* out = (float*)d_out;                  // [B]

    const int B = out_size;                      // 8
    const int N = in_sizes[0] / (B * 3);         // 8192
    const int M = in_sizes[1] / (B * 3);         // 8192

    float* rmP = (float*)d_ws;                   // [B, N]
    float* rmT = rmP + (size_t)B * N;            // [B, M]

    dim3 blk(256);
    dim3 g1((N + ROWS_PER_BLOCK - 1) / ROWS_PER_BLOCK, B);
    chamfer_rowmin_kernel<<<g1, blk, 0, stream>>>(pred, tgt, rmP, N, M);

    dim3 g2((M + ROWS_PER_BLOCK - 1) / ROWS_PER_BLOCK, B);
    chamfer_rowmin_kernel<<<g2, blk, 0, stream>>>(tgt, pred, rmT, M, N);

    chamfer_reduce_kernel<<<B, blk, 0, stream>>>(rmP, rmT, out, N, M);
}